// loTeNet_29076928594611
// MI455X (gfx1250) — compile-verified
//
#include <hip/hip_runtime.h>
#include <hip/hip_bf16.h>

typedef float v2f __attribute__((ext_vector_type(2)));
typedef float v8f __attribute__((ext_vector_type(8)));

#define PI_HALF 1.5707963267948966f

// ---------------------------------------------------------------------------
// MPS layer kernel: one block per pixel n, 256 threads = 8 waves.
// Each wave owns a 16-row batch band, chains V[16,10] through 7 transfer
// matrices using V_WMMA_F32_16X16X4_F32 (K=10 zero-padded to 12 = 3 k-steps).
// Weights are staged in LDS pre-padded to the WMMA tile shape so every
// fragment load is an unconditional DS op (no exec-mask predication).
//   y[b, n] = sum_k V_final[b,k] * r[b,k]
// ---------------------------------------------------------------------------
__global__ __launch_bounds__(256)
void lotenet_mps_layer(const float* __restrict__ x,   // [B, S, S]
                       const float* __restrict__ W0,  // [N, 2, 1, 10]
                       const float* __restrict__ Wm,  // [N, 7, 10, 2, 10]
                       const float* __restrict__ Wn,  // [N, 10, 2]
                       float* __restrict__ y,         // [B, N]
                       int S, int N) {
  __shared__ float WmA[7][12][16];   // A_s = Wm[n,s,:,0,:], zero-padded
  __shared__ float WmB[7][12][16];   // B_s = Wm[n,s,:,1,:], zero-padded
  __shared__ float W0L[20];          // W0[n] : [2][10]
  __shared__ float WnL[20];          // Wn[n] : [10][2]
  __shared__ float csL[8][16][9];    // cos embeddings per (wave,row,site)
  __shared__ float snL[8][16][9];
  __shared__ float Vt[8][16][17];    // wave-private V tile (cols 10..15 = 0)

  const int n    = blockIdx.x;
  const int tid  = threadIdx.x;
  const int wave = tid >> 5;
  const int lane = tid & 31;
  const int b0   = wave * 16;

  // Stage this pixel's transfer matrices into padded LDS tiles (cooperative).
  for (int i = tid; i < 7 * 12 * 16; i += 256) {
    const int s = i / 192, rem = i % 192, k = rem / 16, c = rem % 16;
    float a = 0.f, b = 0.f;
    if (k < 10 && c < 10) {
      const float* p = Wm + (size_t)n * 1400 + ((s * 10 + k) * 2) * 10 + c;
      a = p[0];        // p = 0 plane
      b = p[10];       // p = 1 plane
    }
    WmA[s][k][c] = a;
    WmB[s][k][c] = b;
  }
  if (tid < 20)            W0L[tid]      = W0[(size_t)n * 20 + tid];
  else if (tid < 40)       WnL[tid - 20] = Wn[(size_t)n * 20 + (tid - 20)];

  // Trig embedding for the 9 patch sites of each batch row.
  const int h  = S / 3;
  const int pi = n / h, pj = n % h;
  if (lane < 16) {
    const int b = b0 + lane;
    const float* xb = x + (size_t)b * S * S;
#pragma unroll
    for (int k = 0; k < 9; ++k) {
      float f = xb[(3 * pi + k / 3) * S + (3 * pj + k % 3)];
      float a = PI_HALF * f;
      csL[wave][lane][k] = __cosf(a);
      snL[wave][lane][k] = __sinf(a);
    }
  }

  // Initial V[m, a] = cos0*W0[0,a] + sin0*W0[1,a]   (O == 1)
  __syncthreads();
  if (lane < 16) {
    float c = csL[wave][lane][0], s = snL[wave][lane][0];
#pragma unroll
    for (int a = 0; a < 10; ++a)
      Vt[wave][lane][a] = c * W0L[a] + s * W0L[10 + a];
#pragma unroll
    for (int a = 10; a < 16; ++a) Vt[wave][lane][a] = 0.f;
  }
  __builtin_amdgcn_wave_barrier();

  const int row   = lane & 15;          // A-frag M row / B-frag N column
  const int khalf = (lane >> 4) * 2;    // K sub-offset per lane half
  const int col   = lane & 15;          // C/D column
  const int mbase = (lane >> 4) * 8;    // C/D row group

  for (int s = 0; s < 7; ++s) {
    v8f accA = {};   // V @ A_s
    v8f accB = {};   // V @ B_s
#pragma unroll
    for (int kb = 0; kb < 12; kb += 4) {
      const int k0 = kb + khalf;        // k0+1 <= 11: always in-bounds
      v2f a, bA, bB;
      a.x  = Vt[wave][row][k0];
      a.y  = Vt[wave][row][k0 + 1];
      bA.x = WmA[s][k0][row];
      bA.y = WmA[s][k0 + 1][row];
      bB.x = WmB[s][k0][row];
      bB.y = WmB[s][k0 + 1][row];
      accA = __builtin_amdgcn_wmma_f32_16x16x4_f32(false, a, false, bA,
                                                   (short)0, accA, false, false);
      accB = __builtin_amdgcn_wmma_f32_16x16x4_f32(false, a, false, bB,
                                                   (short)0, accB, false, false);
    }
    // Wave-private tile + per-wave in-order DS: no block barrier needed.
    __builtin_amdgcn_wave_barrier();
    const int site = s + 1;
#pragma unroll
    for (int v = 0; v < 8; ++v) {
      const int m = mbase + v;
      // Cols >= 10 of accA/accB are zero (padded B cols), so V pad stays 0.
      Vt[wave][m][col] = csL[wave][m][site] * accA[v] + snL[wave][m][site] * accB[v];
    }
    __builtin_amdgcn_wave_barrier();
  }

  // Contract with right boundary vector r[a] = cos8*Wn[a,0] + sin8*Wn[a,1].
  if (lane < 16) {
    const int b = b0 + lane;
    float c = csL[wave][lane][8], s = snL[wave][lane][8];
    float acc = 0.f;
#pragma unroll
    for (int k = 0; k < 10; ++k) {
      float r = c * WnL[2 * k] + s * WnL[2 * k + 1];
      acc += Vt[wave][lane][k] * r;
    }
    y[(size_t)b * N + n] = acc;
  }
}

// ---------------------------------------------------------------------------
// BatchNorm statistics: single block reduces sum / sum-of-squares over B*N,
// emits scale = g * rsqrt(var + eps), shift = b - mu*scale.
// ---------------------------------------------------------------------------
__global__ __launch_bounds__(256)
void lotenet_bn_stats(const float* __restrict__ y, int count,
                      const float* __restrict__ gp, const float* __restrict__ bp,
                      float* __restrict__ stats) {
  __shared__ float s1[256], s2[256];
  const int tid = threadIdx.x;
  float a = 0.f, q = 0.f;
  for (int i = tid; i < count; i += 256) {
    float v = y[i];
    a += v;
    q += v * v;
  }
  s1[tid] = a; s2[tid] = q;
  __syncthreads();
  for (int o = 128; o > 0; o >>= 1) {
    if (tid < o) { s1[tid] += s1[tid + o]; s2[tid] += s2[tid + o]; }
    __syncthreads();
  }
  if (tid == 0) {
    float inv_n = 1.f / (float)count;
    float mu  = s1[0] * inv_n;
    float var = s2[0] * inv_n - mu * mu;
    float scale = gp[0] * rsqrtf(var + 1e-5f);
    stats[0] = scale;
    stats[1] = bp[0] - mu * scale;
  }
}

// ---------------------------------------------------------------------------
// Apply BN affine then per-sample min-max normalization -> next layer image.
// One block per batch sample.
// ---------------------------------------------------------------------------
__global__ __launch_bounds__(256)
void lotenet_minmax_norm(const float* __restrict__ y, const float* __restrict__ stats,
                         float* __restrict__ xnext, int N) {
  __shared__ float smn[256], smx[256];
  const int b = blockIdx.x, tid = threadIdx.x;
  const float scale = stats[0], shift = stats[1];
  float mn = 3.402823e38f, mx = -3.402823e38f;
  for (int i = tid; i < N; i += 256) {
    float z = y[(size_t)b * N + i] * scale + shift;
    mn = fminf(mn, z);
    mx = fmaxf(mx, z);
  }
  smn[tid] = mn; smx[tid] = mx;
  __syncthreads();
  for (int o = 128; o > 0; o >>= 1) {
    if (tid < o) {
      smn[tid] = fminf(smn[tid], smn[tid + o]);
      smx[tid] = fmaxf(smx[tid], smx[tid + o]);
    }
    __syncthreads();
  }
  const float base = smn[0];
  const float inv  = 1.f / (smx[0] - base);
  for (int i = tid; i < N; i += 256) {
    float z = y[(size_t)b * N + i] * scale + shift;
    xnext[(size_t)b * N + i] = (z - base) * inv;
  }
}

// ---------------------------------------------------------------------------
// Final MPS head: N=1, O=10. Rows of the O x b matrix evolve independently,
// so one thread handles one (batch, output) row: ~2 KFLOP each, 1280 threads.
// ---------------------------------------------------------------------------
__global__ __launch_bounds__(256)
void lotenet_mps_head(const float* __restrict__ x3,  // [B, 9]
                      const float* __restrict__ W0,  // [1, 2, 10, 10]
                      const float* __restrict__ Wm,  // [1, 7, 10, 2, 10]
                      const float* __restrict__ Wn,  // [1, 10, 2]
                      float* __restrict__ out,       // [B, 10]
                      int B) {
  const int t = blockIdx.x * 256 + threadIdx.x;
  if (t >= B * 10) return;
  const int b = t / 10, o = t % 10;

  float cs[9], sn[9];
#pragma unroll
  for (int k = 0; k < 9; ++k) {
    float a = PI_HALF * x3[b * 9 + k];
    cs[k] = __cosf(a);
    sn[k] = __sinf(a);
  }

  float v[10];
#pragma unroll
  for (int a = 0; a < 10; ++a)
    v[a] = cs[0] * W0[o * 10 + a] + sn[0] * W0[100 + o * 10 + a];

  for (int s = 0; s < 7; ++s) {
    float nv[10];
#pragma unroll
    for (int c = 0; c < 10; ++c) {
      float tA = 0.f, tB = 0.f;
#pragma unroll
      for (int a = 0; a < 10; ++a) {
        tA += v[a] * Wm[((s * 10 + a) * 2 + 0) * 10 + c];
        tB += v[a] * Wm[((s * 10 + a) * 2 + 1) * 10 + c];
      }
      nv[c] = cs[s + 1] * tA + sn[s + 1] * tB;
    }
#pragma unroll
    for (int c = 0; c < 10; ++c) v[c] = nv[c];
  }

  float acc = 0.f;
#pragma unroll
  for (int a = 0; a < 10; ++a) {
    float r = cs[8] * Wn[a * 2] + sn[8] * Wn[a * 2 + 1];
    acc += v[a] * r;
  }
  out[b * 10 + o] = acc;
}

// ---------------------------------------------------------------------------
extern "C" void kernel_launch(void* const* d_in, const int* in_sizes, int n_in,
                              void* d_out, int out_size, void* d_ws, size_t ws_size,
                              hipStream_t stream) {
  (void)in_sizes; (void)n_in; (void)out_size; (void)ws_size;
  const int B = 128;

  const float* x    = (const float*)d_in[0];
  const float* W0_0 = (const float*)d_in[1];
  const float* Wm_0 = (const float*)d_in[2];
  const float* Wn_0 = (const float*)d_in[3];
  const float* g_0  = (const float*)d_in[4];
  const float* b_0  = (const float*)d_in[5];
  const float* W0_1 = (const float*)d_in[6];
  const float* Wm_1 = (const float*)d_in[7];
  const float* Wn_1 = (const float*)d_in[8];
  const float* g_1  = (const float*)d_in[9];
  const float* b_1  = (const float*)d_in[10];
  const float* W0_2 = (const float*)d_in[11];
  const float* Wm_2 = (const float*)d_in[12];
  const float* Wn_2 = (const float*)d_in[13];
  const float* g_2  = (const float*)d_in[14];
  const float* b_2  = (const float*)d_in[15];
  const float* W0_3 = (const float*)d_in[16];
  const float* Wm_3 = (const float*)d_in[17];
  const float* Wn_3 = (const float*)d_in[18];
  float* out = (float*)d_out;

  // Workspace layout (floats), 256-float aligned slices. Total < 1 MB.
  float* ws = (float*)d_ws;
  size_t off = 0;
  float* y0 = ws + off; off += (size_t)B * 729; off = (off + 255) & ~255ull;
  float* x1 = ws + off; off += (size_t)B * 729; off = (off + 255) & ~255ull;
  float* y1 = ws + off; off += (size_t)B * 81;  off = (off + 255) & ~255ull;
  float* x2 = ws + off; off += (size_t)B * 81;  off = (off + 255) & ~255ull;
  float* y2 = ws + off; off += (size_t)B * 9;   off = (off + 255) & ~255ull;
  float* x3 = ws + off; off += (size_t)B * 9;   off = (off + 255) & ~255ull;
  float* stats = ws + off;

  // Layer 0: 81x81 image, N = 729 patches
  lotenet_mps_layer<<<729, 256, 0, stream>>>(x,  W0_0, Wm_0, Wn_0, y0, 81, 729);
  lotenet_bn_stats<<<1, 256, 0, stream>>>(y0, B * 729, g_0, b_0, stats);
  lotenet_minmax_norm<<<B, 256, 0, stream>>>(y0, stats, x1, 729);

  // Layer 1: 27x27 image, N = 81
  lotenet_mps_layer<<<81, 256, 0, stream>>>(x1, W0_1, Wm_1, Wn_1, y1, 27, 81);
  lotenet_bn_stats<<<1, 256, 0, stream>>>(y1, B * 81, g_1, b_1, stats);
  lotenet_minmax_norm<<<B, 256, 0, stream>>>(y1, stats, x2, 81);

  // Layer 2: 9x9 image, N = 9
  lotenet_mps_layer<<<9, 256, 0, stream>>>(x2, W0_2, Wm_2, Wn_2, y2, 9, 9);
  lotenet_bn_stats<<<1, 256, 0, stream>>>(y2, B * 9, g_2, b_2, stats);
  lotenet_minmax_norm<<<B, 256, 0, stream>>>(y2, stats, x3, 9);

  // Head: 3x3 image -> [B, 10]
  lotenet_mps_head<<<(B * 10 + 255) / 256, 256, 0, stream>>>(x3, W0_3, Wm_3, Wn_3, out, B);
}